// Attention_74113955660046
// MI455X (gfx1250) — compile-verified
//
#include <hip/hip_runtime.h>
#include <stdint.h>

// Problem constants (from the reference): B=64, T=4096, D=512, fp32.
#define NB    64
#define NT    4096
#define ND    512
#define TR    32                         // tile rows staged in LDS per step
#define NTHR  256                        // 8 waves (wave32)
#define TILE_BYTES (TR * ND * 4)         // 64 KiB per tile buffer
#define UNITS (TILE_BYTES / (16 * NTHR)) // 16 b128 async issues per thread per tile
#define WS_STRIDE 520                    // floats per partial record: [s][pad...][acc 512]

typedef __attribute__((ext_vector_type(2))) float v2f;
typedef __attribute__((ext_vector_type(8))) float v8f;

// ---- gfx1250 tanh (native v_tanh_f32 if available) ------------------------
#ifdef __has_builtin
#if __has_builtin(__builtin_amdgcn_tanhf)
#define TANHF(v) __builtin_amdgcn_tanhf(v)
#endif
#endif
#ifndef TANHF
#define TANHF(v) tanhf(v)
#endif

// ---- CDNA5 async global->LDS copy (ASYNCcnt-tracked) ----------------------
// vdst = VGPR holding LDS byte offset (low 32 bits of the generic shared ptr,
// which per the gfx1250 aperture rules ARE the wave-relative LDS address).
// x is streamed exactly once and lands in LDS -> non-temporal on the global
// side so 512 MiB of stream traffic does not thrash the 192 MB L2.
__device__ __forceinline__ void async_ld_b128(uint32_t lds_off, const char* g) {
  asm volatile("global_load_async_to_lds_b128 %0, %1, off th:TH_LOAD_NT"
               :: "v"(lds_off), "v"(g) : "memory");
}
__device__ __forceinline__ void wait_async_le16() {
  asm volatile("s_wait_asynccnt 0x10" ::: "memory");
}
__device__ __forceinline__ void wait_async_0() {
  asm volatile("s_wait_asynccnt 0x0" ::: "memory");
}

// Issue one 64 KiB tile: 256 threads x 16 x b128 (each wave issues 16 async ops,
// so two tiles in flight = 32 outstanding, well under the 6-bit ASYNCcnt).
__device__ __forceinline__ void issue_tile(uint32_t lds_off, const char* g, int tid) {
  const char* gp = g + tid * 16;
  uint32_t lp = lds_off + (uint32_t)(tid * 16);
#pragma unroll
  for (int j = 0; j < UNITS; ++j) {
    async_ld_b128(lp, gp);
    gp += 16 * NTHR;
    lp += 16 * NTHR;
  }
}

__device__ __forceinline__ float dot4(float4 a, float4 b, float acc) {
  acc = fmaf(a.x, b.x, acc); acc = fmaf(a.y, b.y, acc);
  acc = fmaf(a.z, b.z, acc); acc = fmaf(a.w, b.w, acc);
  return acc;
}
__device__ __forceinline__ void fma4(float4& a, float p, float4 xv) {
  a.x = fmaf(p, xv.x, a.x); a.y = fmaf(p, xv.y, a.y);
  a.z = fmaf(p, xv.z, a.z); a.w = fmaf(p, xv.w, a.w);
}

// Main streaming kernel: one block handles rows [chunk*(T/S), (chunk+1)*(T/S))
// of one batch. Single pass over x: tanh bounds e to [-1,1], so exp(e) is in
// [0.37, 2.72] and can never overflow -> no online-softmax rescaling needed.
// S and the output mode are compile-time so the prologue is shift/mask only.
template <int S, bool WRITE_FINAL>
__global__ __launch_bounds__(NTHR)
void attn_pool_main(const float* __restrict__ x,
                    const float* __restrict__ Wp,
                    const float* __restrict__ bp,
                    float* __restrict__ dst)   // ws partials or final out
{
  extern __shared__ char smem[];               // 2 * TILE_BYTES
  constexpr int CH     = NT / S;               // rows per chunk
  constexpr int NTILES = CH / TR;              // 8 (S=16) .. 128 (S=1), >= 2
  const int tid  = threadIdx.x;
  const int lane = tid & 31;
  const int wv   = tid >> 5;                   // wave id, 0..7
  const int batch = (int)blockIdx.x / S;       // S is a power of two -> shift
  const int chunk = (int)blockIdx.x - batch * S;
  const int t0    = chunk * CH;

  const char* gx = (const char*)(x + (size_t)batch * NT * ND + (size_t)t0 * ND);
  const uint32_t lds_base = (uint32_t)(uintptr_t)&smem[0];

  // Each lane owns d-slice [lane*16, lane*16+16): keep its W slice in regs.
  const float4* W4 = (const float4*)Wp;
  float4 w0 = W4[lane * 4 + 0], w1 = W4[lane * 4 + 1];
  float4 w2 = W4[lane * 4 + 2], w3 = W4[lane * 4 + 3];

  float  s_sum = 0.f;
  float4 a0 = make_float4(0.f, 0.f, 0.f, 0.f), a1 = a0, a2 = a0, a3 = a0;

  // Prime the double buffer (NTILES >= 2 always).
  issue_tile(lds_base,              gx,              tid);
  issue_tile(lds_base + TILE_BYTES, gx + TILE_BYTES, tid);

  for (int i = 0; i < NTILES; ++i) {
    // Oldest 16 async ops (tile i) done for this wave; barrier => for all waves.
    if (i + 1 < NTILES) wait_async_le16(); else wait_async_0();
    __syncthreads();

    const float* buf = (const float*)(smem + (size_t)(i & 1) * TILE_BYTES);
#pragma unroll
    for (int k = 0; k < TR / 8; ++k) {         // 4 rows per wave per tile
      const int r = wv + 8 * k;
      const float4* row = (const float4*)(buf + r * ND);
      float4 x0 = row[lane * 4 + 0], x1 = row[lane * 4 + 1];
      float4 x2 = row[lane * 4 + 2], x3 = row[lane * 4 + 3];

      // partial dot over this lane's 16 d's (x stays in regs for the acc FMAs)
      float p = 0.f;
      p = dot4(x0, w0, p); p = dot4(x1, w1, p);
      p = dot4(x2, w2, p); p = dot4(x3, w3, p);
      // full-wave butterfly: every lane ends with the complete dot
#pragma unroll
      for (int off = 16; off > 0; off >>= 1) p += __shfl_xor(p, off, 32);

      const float bt = bp[t0 + i * TR + r];    // uniform per wave -> scalar load
      const float e  = TANHF(p + bt);
      const float pe = __expf(e);              // safe: e in [-1,1]
      s_sum += pe;
      fma4(a0, pe, x0); fma4(a1, pe, x1); fma4(a2, pe, x2); fma4(a3, pe, x3);
    }
    __syncthreads();                            // all waves done reading buf
    if (i + 2 < NTILES)
      issue_tile(lds_base + (uint32_t)((i & 1) * TILE_BYTES),
                 gx + (size_t)(i + 2) * TILE_BYTES, tid);
  }

  // ---- cross-wave merge (reuse buffer 0: all async + compute are done) ----
  float* accS = (float*)smem;                  // 8 waves * 512 floats = 16 KiB
  float* sS   = (float*)(smem + 8 * ND * 4);   // 8 floats
  float4* myA = (float4*)(accS + (size_t)wv * ND);
  myA[lane * 4 + 0] = a0; myA[lane * 4 + 1] = a1;
  myA[lane * 4 + 2] = a2; myA[lane * 4 + 3] = a3;
  if (lane == 0) sS[wv] = s_sum;               // s_sum identical across lanes
  __syncthreads();

  float stot = 0.f;
#pragma unroll
  for (int ww = 0; ww < 8; ++ww) stot += sS[ww];

  // The merge is ones[1x8] x accS[8x512]: do it on the matrix pipe with full
  // fp32 WMMA (16x16x4). Each wave sums all 8 partials for 64 columns:
  // per 16-col tile, two chained V_WMMA_F32_16X16X4_F32 reduce rows 0-3 and
  // 4-7; with A==1 every C row equals the column sum, so c[0] holds the
  // answer for this lane's column (broadcast for free, no shuffles).
  // B layout (4x16 f32): lane<16 -> {B[0][n],B[1][n]}, lane>=16 -> {B[2][n],B[3][n]}.
  const int half = lane >> 4;                  // 0: rows {0,1}/{4,5}; 1: rows {2,3}/{6,7}
  const int nc   = lane & 15;                  // column within the 16-col tile
  v2f ones; ones[0] = 1.f; ones[1] = 1.f;      // A matrix: all ones
  float colsum[4];
#pragma unroll
  for (int ct = 0; ct < 4; ++ct) {
    const int cb = wv * 64 + ct * 16;          // this wave's column-tile base
    v2f b0, b1;
    b0[0] = accS[(2 * half + 0) * ND + cb + nc];
    b0[1] = accS[(2 * half + 1) * ND + cb + nc];
    b1[0] = accS[(4 + 2 * half + 0) * ND + cb + nc];
    b1[1] = accS[(4 + 2 * half + 1) * ND + cb + nc];
    v8f c = {};
    c = __builtin_amdgcn_wmma_f32_16x16x4_f32(false, ones, false, b0,
                                              (short)0, c, false, false);
    c = __builtin_amdgcn_wmma_f32_16x16x4_f32(false, ones, false, b1,
                                              (short)0, c, false, false);
    colsum[ct] = c[0];
  }

  if (WRITE_FINAL) {                           // S==1 path: direct output
    const float inv = 1.0f / stot;
    if (half == 0) {
#pragma unroll
      for (int ct = 0; ct < 4; ++ct)
        dst[(size_t)batch * ND + wv * 64 + ct * 16 + nc] = colsum[ct] * inv;
    }
  } else {                                     // per-chunk partial to workspace
    float* wsf = dst + (size_t)blockIdx.x * WS_STRIDE;
    if (tid == 0) wsf[0] = stot;
    if (half == 0) {
#pragma unroll
      for (int ct = 0; ct < 4; ++ct)
        wsf[8 + wv * 64 + ct * 16 + nc] = colsum[ct];
    }
  }
}

// Tiny epilogue: combine S chunk-partials per batch, divide, write out.
// S is compile-time -> fully unrolled constant-offset loads, no division math.
template <int S>
__global__ __launch_bounds__(256)
void attn_pool_combine(const float* __restrict__ ws, float* __restrict__ out)
{
  const int batch = blockIdx.x, tid = threadIdx.x;
  const float* base = ws + (size_t)batch * S * WS_STRIDE;
  float stot = 0.f, r0 = 0.f, r1 = 0.f;
#pragma unroll
  for (int c = 0; c < S; ++c) {
    const float* p = base + (size_t)c * WS_STRIDE;
    stot += p[0];
    r0   += p[8 + tid];
    r1   += p[8 + tid + 256];
  }
  const float inv = 1.0f / stot;
  out[(size_t)batch * ND + tid]       = r0 * inv;
  out[(size_t)batch * ND + tid + 256] = r1 * inv;
}

extern "C" void kernel_launch(void* const* d_in, const int* in_sizes, int n_in,
                              void* d_out, int out_size, void* d_ws, size_t ws_size,
                              hipStream_t stream)
{
  const float* x = (const float*)d_in[0];   // [64, 4096, 512] fp32
  const float* W = (const float*)d_in[1];   // [512, 1]       fp32
  const float* b = (const float*)d_in[2];   // [4096, 1]      fp32
  float* out = (float*)d_out;               // [64, 512]      fp32

  const size_t shmem = 2 * TILE_BYTES;      // 128 KiB: 2 WGs fit per 320 KiB WGP
  const size_t need16 = (size_t)NB * 16 * WS_STRIDE * sizeof(float); // ~2.1 MB
  const size_t need8  = (size_t)NB *  8 * WS_STRIDE * sizeof(float); // ~1.06 MB

  if (d_ws != nullptr && ws_size >= need16) {
    attn_pool_main<16, false><<<NB * 16, NTHR, shmem, stream>>>(x, W, b, (float*)d_ws);
    attn_pool_combine<16><<<NB, 256, 0, stream>>>((const float*)d_ws, out);
  } else if (d_ws != nullptr && ws_size >= need8) {
    attn_pool_main<8, false><<<NB * 8, NTHR, shmem, stream>>>(x, W, b, (float*)d_ws);
    attn_pool_combine<8><<<NB, 256, 0, stream>>>((const float*)d_ws, out);
  } else {
    attn_pool_main<1, true><<<NB, NTHR, shmem, stream>>>(x, W, b, out);
  }
}